// DefAddkeysTransformer_21363167330740
// MI455X (gfx1250) — compile-verified
//
#include <hip/hip_runtime.h>
#include <hip/hip_bf16.h>
#include <math.h>
#include <stdint.h>

typedef float v2f __attribute__((ext_vector_type(2)));
typedef float v8f __attribute__((ext_vector_type(8)));

// Problem constants
constexpr int DM    = 256;    // d_model
constexpr int LQ    = 1024;   // queries
constexpr int NHD   = 8;      // heads
constexpr int NLV   = 4;      // levels
constexpr int NPT   = 4;      // points
constexpr int NLOG  = NLV*NPT + 256;  // 272 logit rows per head

// Workspace byte offsets
constexpr size_t WS_IDX    = 0;                                   // int[LQ*NHD*NLV*NPT]
constexpr size_t WS_LOGITS = 524288;                              // f32[NHD*NLOG*LQ]
constexpr size_t WS_KI2    = WS_LOGITS + 8912896;                 // f32[NHD*256*256]
constexpr size_t WS_ASUM   = WS_KI2 + 2097152;                    // f32[NHD*LQ]
constexpr size_t WS_U      = WS_ASUM + 32768;                     // f32[NHD*LQ*256]
constexpr size_t WS_V2     = WS_U + 8388608;                      // f32[NHD*256*256]
constexpr size_t WS_HW     = WS_V2 + 2097152;                     // f32[9*256]

__device__ __forceinline__ int lane_id() { return threadIdx.x & 31; }

__device__ __forceinline__ v8f wmma4(v2f a, v2f b, v8f c) {
  // D = A(16x4,f32) * B(4x16,f32) + C(16x16,f32)
  return __builtin_amdgcn_wmma_f32_16x16x4_f32(false, a, false, b, (short)0, c, false, false);
}

// A tile 16x4: A[m][k] = src[m*ld + k0+k]   (lanes 0-15: K=0,1 ; 16-31: K=2,3)
__device__ __forceinline__ v2f ldA(const float* src, int ld, int k0) {
  int l = lane_id(); int m = l & 15; int ks = (l >> 4) << 1;
  v2f r; r.x = src[m*ld + k0 + ks]; r.y = src[m*ld + k0 + ks + 1]; return r;
}
// A tile 16x4, K-major source: A[m][k] = src[(k0+k)*ld + m]
__device__ __forceinline__ v2f ldA_km(const float* src, int ld, int k0) {
  int l = lane_id(); int m = l & 15; int ks = (l >> 4) << 1;
  v2f r; r.x = src[(k0+ks)*ld + m]; r.y = src[(k0+ks+1)*ld + m]; return r;
}
// B tile 4x16, K-major: B[k][n] = src[(k0+k)*ld + n]
__device__ __forceinline__ v2f ldB_km(const float* src, int ld, int k0) {
  int l = lane_id(); int n = l & 15; int ks = (l >> 4) << 1;
  v2f r; r.x = src[(k0+ks)*ld + n]; r.y = src[(k0+ks+1)*ld + n]; return r;
}
// B tile 4x16, N-major (transposed weight): B[k][n] = src[n*ld + k0+k]
__device__ __forceinline__ v2f ldB_nm(const float* src, int ld, int k0) {
  int l = lane_id(); int n = l & 15; int ks = (l >> 4) << 1;
  v2f r; r.x = src[n*ld + k0 + ks]; r.y = src[n*ld + k0 + ks + 1]; return r;
}

// ---------- kernel 0: head mixing weights: softmax over 9 heads per channel ----------
__global__ __launch_bounds__(256) void k_headw(const float* __restrict__ Wmix,
                                               float* __restrict__ hw) {
  int c = threadIdx.x;
  float v[9]; float m = -1e30f;
  for (int h = 0; h < 9; ++h) { v[h] = Wmix[c*9 + h]; m = fmaxf(m, v[h]); }
  float s = 0.f;
  for (int h = 0; h < 9; ++h) { v[h] = __expf(v[h] - m); s += v[h]; }
  float inv = 1.f / s;
  for (int h = 0; h < 9; ++h) hw[h*256 + c] = v[h] * inv;
}

// ---------- kernel 1: offsets GEMM (WMMA) fused with sampling-index computation ----------
__global__ __launch_bounds__(128) void k_off_idx(const float* __restrict__ query,
                                                 const float* __restrict__ refp,
                                                 const int*   __restrict__ iss,
                                                 const int*   __restrict__ lstart,
                                                 const float* __restrict__ Woff,
                                                 const float* __restrict__ boff,
                                                 int* __restrict__ fidx) {
  __shared__ float S[16*256];
  int q0 = blockIdx.x * 16;
  int w  = threadIdx.x >> 5;
  int l  = lane_id(); int n = l & 15; int hi = l >> 4;

  v8f acc[4]; for (int t = 0; t < 4; ++t) acc[t] = {};
  const float* Aq = query + (size_t)q0 * DM;
  for (int k0 = 0; k0 < DM; k0 += 4) {
    v2f a = ldA(Aq, DM, k0);
    for (int nt = 0; nt < 4; ++nt) {
      const float* Bm = Woff + (size_t)(w*64 + nt*16) * DM;
      v2f b = ldB_nm(Bm, DM, k0);
      acc[nt] = wmma4(a, b, acc[nt]);
    }
  }
  for (int nt = 0; nt < 4; ++nt)
    for (int v = 0; v < 8; ++v) {
      int m = v + hi*8;
      S[m*256 + w*64 + nt*16 + n] = acc[nt][v];
    }
  __syncthreads();

  for (int it = threadIdx.x; it < 16*128; it += 128) {
    int ql = it >> 7, u = it & 127;
    int h = u >> 4, rem = u & 15, lv = rem >> 2, p = rem & 3;
    int q = q0 + ql;
    float off0 = S[ql*256 + 2*u    ] + boff[2*u    ];
    float off1 = S[ql*256 + 2*u + 1] + boff[2*u + 1];
    int   Hs = iss[lv*2 + 0], Ws = iss[lv*2 + 1];
    float Hf = (float)Hs, Wf = (float)Ws;
    float r0 = refp[(q*NLV + lv)*2 + 0];
    float r1 = refp[(q*NLV + lv)*2 + 1];
    float loc0 = fminf(fmaxf(r0 + off0 / Wf, 0.f), 0.999f);
    float loc1 = fminf(fmaxf(r1 + off1 / Hf, 0.f), 0.999f);
    int ix0 = (int)(loc0 * Hf);
    int ix1 = (int)(loc1 * Wf);
    fidx[((q*NHD + h)*NLV + lv)*NPT + p] = ix0 + ix1*Hs + lstart[lv];
  }
}

// ---------- kernel 2: sample-point attention logits ----------
// For (h,l,g): T_g(64x256) = Q_g @ Kb_g  via WMMA, keys streamed into LDS with
// double-buffered GLOBAL_LOAD_ASYNC_TO_LDS_B128 (ASYNCcnt), then
// attn[p][t] = dot(W_attn_row(p*64+t), T_g[t]).
constexpr int LDP = 264;   // padded LDS row stride (floats)

// Issue one 64-row key chunk as per-lane async global->LDS B128 copies.
// Each thread copies 256 contiguous bytes (16 x b128); 16 async ops per wave.
__device__ __forceinline__ void issue_key_chunk_async(const float* __restrict__ inflat,
                                                      const int*   __restrict__ fidx,
                                                      float* Kbuf, int ch,
                                                      int g, int lv, int h, int tid) {
  int rowgrp = tid >> 2;          // 0..63 : chunk-local key row
  int quad   = tid & 3;           // 4 threads per row, 256B each
  int qk = (g & 3)*256 + ch*64 + rowgrp;
  int fr = fidx[((qk*NHD + h)*NLV + lv)*NPT + (g >> 2)];
  uint64_t gbase = (uint64_t)(uintptr_t)(inflat + (size_t)fr*DM) + (uint64_t)(quad*256);
  uint32_t lbase = (uint32_t)(uintptr_t)(Kbuf + rowgrp*LDP) + (uint32_t)(quad*256);
  #pragma unroll
  for (int j = 0; j < 16; ++j) {
    uint32_t la = lbase + j*16;
    uint64_t ga = gbase + j*16;
    asm volatile("global_load_async_to_lds_b128 %0, %1, off"
                 :: "v"(la), "v"(ga) : "memory");
  }
}

__global__ __launch_bounds__(256) void k_samp_logits(const float* __restrict__ query,
                                                     const float* __restrict__ inflat,
                                                     const float* __restrict__ Wattn,
                                                     const int*   __restrict__ fidx,
                                                     float* __restrict__ logits) {
  extern __shared__ float sm[];
  float* Qs = sm;                  // 64 x LDP : query group
  float* K0 = sm + 64*LDP;         // key chunk buffer 0 (reused for T)
  float* K1 = sm + 2*64*LDP;       // key chunk buffer 1
  int g = blockIdx.x, lv = blockIdx.y, h = blockIdx.z;
  int tid = threadIdx.x; int w = tid >> 5;
  int l = tid & 31; int n = l & 15; int hi = l >> 4;

  // Prefetch this thread's W_attn reduction row (used after the GEMM phase).
  int rp = tid >> 6, rt = tid & 63;
  const float* simrow = Wattn + (size_t)(h*4 + lv)*DM*DM + (size_t)(rp*64 + rt)*DM;
  __builtin_prefetch(simrow, 0, 1);
  __builtin_prefetch(simrow + 128, 0, 1);

  // Kick off the first async key chunk immediately.
  issue_key_chunk_async(inflat, fidx, K0, 0, g, lv, h, tid);

  // Stage the 64 queries of this group (q = 16t + g) while chunk 0 streams in.
  for (int idx = tid; idx < 64*64; idx += 256) {
    int t = idx >> 6, c4 = idx & 63;
    int q = 16*t + g;
    ((float4*)(Qs + t*LDP))[c4] = ((const float4*)(query + (size_t)q*DM))[c4];
  }

  int mt = w & 3, nh2 = w >> 2;
  v8f acc[8]; for (int j = 0; j < 8; ++j) acc[j] = {};

  for (int ch = 0; ch < 4; ++ch) {
    float* cur = (ch & 1) ? K1 : K0;
    float* nxt = (ch & 1) ? K0 : K1;
    if (ch < 3) {
      // Stream the next chunk while we compute on the current one.
      issue_key_chunk_async(inflat, fidx, nxt, ch + 1, g, lv, h, tid);
      // In-order completion: <=16 outstanding => current chunk has landed.
      asm volatile("s_wait_asynccnt 16" ::: "memory");
    } else {
      asm volatile("s_wait_asynccnt 0" ::: "memory");
    }
    __syncthreads();

    const float* Arow = Qs + (mt*16)*LDP + ch*64;
    for (int k0 = 0; k0 < 64; k0 += 4) {
      v2f a = ldA(Arow, LDP, k0);
      for (int j = 0; j < 8; ++j) {
        v2f b = ldB_km(cur + nh2*128 + j*16, LDP, k0);
        acc[j] = wmma4(a, b, acc[j]);
      }
    }
    __syncthreads();   // protect 'nxt' (re-issued two iterations later) and 'cur' reuse
  }

  // Stage T into K0 (last compute used K1; K0 reads finished at ch=2 barrier).
  for (int j = 0; j < 8; ++j) {
    int n0 = nh2*128 + j*16;
    for (int v = 0; v < 8; ++v) {
      int m = v + hi*8;
      K0[(mt*16 + m)*LDP + n0 + n] = acc[j][v];
    }
  }
  __syncthreads();

  // attn logit reduction: one (p,t) per thread
  const float* tr = K0 + rt*LDP;
  float s = 0.f;
  #pragma unroll 8
  for (int d = 0; d < DM; ++d) s += simrow[d]*tr[d];
  logits[((size_t)h*NLOG + lv*NPT + rp)*LQ + 16*rt + g] = s;
}

// ---------- kernel 3a: ki2[h] = add_keys @ W_attn[h*4+4]^T ----------
__global__ __launch_bounds__(128) void k_ki2(const float* __restrict__ addk,
                                             const float* __restrict__ Wattn,
                                             float* __restrict__ ki2) {
  int h = blockIdx.z; int j0 = blockIdx.x*16; int w = threadIdx.x >> 5;
  int c0 = (blockIdx.y*4 + w)*16;
  int l = lane_id(); int n = l & 15; int hi = l >> 4;
  const float* A  = addk + (size_t)j0*DM;
  const float* Bm = Wattn + (size_t)(h*4 + 4)*DM*DM + (size_t)c0*DM;
  v8f acc = {};
  for (int k0 = 0; k0 < DM; k0 += 4)
    acc = wmma4(ldA(A, DM, k0), ldB_nm(Bm, DM, k0), acc);
  float* Dr = ki2 + (size_t)h*DM*DM + (size_t)j0*DM + c0;
  for (int v = 0; v < 8; ++v) Dr[(v + hi*8)*DM + n] = acc[v];
}

// ---------- kernel 3b: add-key logits = ki2[h] @ query^T ----------
__global__ __launch_bounds__(128) void k_add_logits(const float* __restrict__ ki2,
                                                    const float* __restrict__ query,
                                                    float* __restrict__ logits) {
  int h = blockIdx.z; int j0 = blockIdx.x*16; int w = threadIdx.x >> 5;
  int q0 = (blockIdx.y*4 + w)*16;
  int l = lane_id(); int n = l & 15; int hi = l >> 4;
  const float* A  = ki2 + (size_t)h*DM*DM + (size_t)j0*DM;
  const float* Bq = query + (size_t)q0*DM;   // B[k][n] = query[q0+n][k]
  v8f acc = {};
  for (int k0 = 0; k0 < DM; k0 += 4)
    acc = wmma4(ldA(A, DM, k0), ldB_nm(Bq, DM, k0), acc);
  for (int v = 0; v < 8; ++v) {
    int m = v + hi*8;
    logits[((size_t)h*NLOG + 16 + j0 + m)*LQ + q0 + n] = acc[v];
  }
}

// ---------- kernel 4: column softmax over 272 logits per (h,q); asum of first 16 ----------
__global__ __launch_bounds__(256) void k_softmax(float* __restrict__ logits,
                                                 float* __restrict__ asum) {
  int col = blockIdx.x*256 + threadIdx.x;     // 8192 columns
  int h = col >> 10, q = col & 1023;
  float* base = logits + (size_t)h*NLOG*LQ + q;
  float m = -1e30f;
  for (int i = 0; i < NLOG; ++i) m = fmaxf(m, base[(size_t)i*LQ]);
  float s = 0.f;
  for (int i = 0; i < NLOG; ++i) s += __expf(base[(size_t)i*LQ] - m);
  float inv = 1.f / s, a16 = 0.f;
  for (int i = 0; i < NLOG; ++i) {
    float e = __expf(base[(size_t)i*LQ] - m) * inv;
    base[(size_t)i*LQ] = e;
    if (i < 16) a16 += e;
  }
  asum[col] = a16;
}

// ---------- kernel 5: U[h][q] = sum_{i<16} a_i * gathered key row ----------
__global__ __launch_bounds__(256) void k_U(const float* __restrict__ logits,
                                           const int*   __restrict__ fidx,
                                           const float* __restrict__ inflat,
                                           float* __restrict__ U) {
  int h = blockIdx.y; int qbase = blockIdx.x*64;
  __shared__ float aw[16]; __shared__ int rows[16];
  int tid = threadIdx.x;
  for (int qq = 0; qq < 64; ++qq) {
    int q = qbase + qq;
    if (tid < 16) {
      aw[tid]   = logits[((size_t)h*NLOG + tid)*LQ + q];
      rows[tid] = fidx[((q*NHD + h)*NLV + (tid >> 2))*NPT + (tid & 3)];
      // warm the gathered key rows for the whole block
      __builtin_prefetch(inflat + (size_t)rows[tid]*DM, 0, 0);
    }
    __syncthreads();
    float acc = 0.f;
    #pragma unroll
    for (int i = 0; i < 16; ++i) acc += aw[i] * inflat[(size_t)rows[i]*DM + tid];
    U[((size_t)h*LQ + q)*DM + tid] = acc;
    __syncthreads();
  }
}

// ---------- kernel 6: v2[h] = add_keys @ W_val[2h+1]^T + b_val[2h+1] ----------
__global__ __launch_bounds__(128) void k_v2(const float* __restrict__ addk,
                                            const float* __restrict__ Wval,
                                            const float* __restrict__ bval,
                                            float* __restrict__ v2w) {
  int h = blockIdx.z; int j0 = blockIdx.x*16; int w = threadIdx.x >> 5;
  int c0 = (blockIdx.y*4 + w)*16;
  int l = lane_id(); int n = l & 15; int hi = l >> 4;
  const float* A  = addk + (size_t)j0*DM;
  const float* Bm = Wval + (size_t)(2*h + 1)*DM*DM + (size_t)c0*DM;
  v8f acc = {};
  for (int k0 = 0; k0 < DM; k0 += 4)
    acc = wmma4(ldA(A, DM, k0), ldB_nm(Bm, DM, k0), acc);
  float bb = bval[(2*h + 1)*DM + c0 + n];
  float* Dr = v2w + (size_t)h*DM*DM + (size_t)j0*DM + c0;
  for (int v = 0; v < 8; ++v) Dr[(v + hi*8)*DM + n] = acc[v] + bb;
}

// ---------- kernel 7: final fused dual-GEMM + head mixing ----------
__global__ __launch_bounds__(128) void k_final(const float* __restrict__ query,
                                               const float* __restrict__ Wval,
                                               const float* __restrict__ bval,
                                               const float* __restrict__ logits,
                                               const float* __restrict__ asum,
                                               const float* __restrict__ U,
                                               const float* __restrict__ v2w,
                                               const float* __restrict__ hw,
                                               float* __restrict__ out) {
  int q0 = blockIdx.x*16;
  int w  = threadIdx.x >> 5;
  int c0 = (blockIdx.y*4 + w)*16;
  int l = lane_id(); int n = l & 15; int hi = l >> 4;
  v8f tot = {};
  for (int h = 0; h < NHD; ++h) {
    v8f acc = {};
    const float* Ub = U    + ((size_t)h*LQ + q0)*DM;
    const float* Wv = Wval + (size_t)(2*h)*DM*DM + (size_t)c0*DM;
    const float* Lg = logits + ((size_t)h*NLOG + 16)*LQ + q0;
    const float* V2 = v2w  + (size_t)h*DM*DM + c0;
    for (int k0 = 0; k0 < DM; k0 += 4) {
      acc = wmma4(ldA(Ub, DM, k0),  ldB_nm(Wv, DM, k0), acc);   // U @ Wv^T
      acc = wmma4(ldA_km(Lg, LQ, k0), ldB_km(V2, DM, k0), acc); // a2^T @ v2
    }
    float hwn = hw[h*256 + c0 + n];
    float bvn = bval[2*h*DM + c0 + n];
    for (int v = 0; v < 8; ++v) {
      int m = v + hi*8;
      float as = asum[(size_t)h*LQ + q0 + m];
      tot[v] += hwn * (acc[v] + as * bvn);
    }
  }
  float hw8 = hw[8*256 + c0 + n];
  for (int v = 0; v < 8; ++v) {
    int m = v + hi*8;
    out[(size_t)(q0 + m)*DM + c0 + n] =
        tot[v] + query[(size_t)(q0 + m)*DM + c0 + n] * hw8;
  }
}

extern "C" void kernel_launch(void* const* d_in, const int* in_sizes, int n_in,
                              void* d_out, int out_size, void* d_ws, size_t ws_size,
                              hipStream_t stream) {
  const float* query = (const float*)d_in[0];
  const float* refp  = (const float*)d_in[1];
  const float* inflat= (const float*)d_in[2];
  const int*   iss   = (const int*)  d_in[3];
  const float* addk  = (const float*)d_in[4];
  const int*   lst   = (const int*)  d_in[5];
  const float* Woff  = (const float*)d_in[6];
  const float* boff  = (const float*)d_in[7];
  const float* Wattn = (const float*)d_in[8];
  const float* Wval  = (const float*)d_in[9];
  const float* bval  = (const float*)d_in[10];
  const float* Wmix  = (const float*)d_in[11];
  float* out = (float*)d_out;

  char* ws = (char*)d_ws;
  int*   fidx   = (int*)  (ws + WS_IDX);
  float* logits = (float*)(ws + WS_LOGITS);
  float* ki2    = (float*)(ws + WS_KI2);
  float* asum   = (float*)(ws + WS_ASUM);
  float* Ubuf   = (float*)(ws + WS_U);
  float* v2w    = (float*)(ws + WS_V2);
  float* hw     = (float*)(ws + WS_HW);

  k_headw  <<<1, 256, 0, stream>>>(Wmix, hw);
  k_off_idx<<<64, 128, 0, stream>>>(query, refp, iss, lst, Woff, boff, fidx);
  k_samp_logits<<<dim3(16,4,8), 256, 3*64*LDP*sizeof(float), stream>>>(query, inflat, Wattn, fidx, logits);
  k_ki2       <<<dim3(16,4,8), 128, 0, stream>>>(addk, Wattn, ki2);
  k_add_logits<<<dim3(16,16,8),128, 0, stream>>>(ki2, query, logits);
  k_softmax   <<<32, 256, 0, stream>>>(logits, asum);
  k_U         <<<dim3(16,8), 256, 0, stream>>>(logits, fidx, inflat, Ubuf);
  k_v2        <<<dim3(16,4,8), 128, 0, stream>>>(addk, Wval, bval, v2w);
  k_final     <<<dim3(64,4), 128, 0, stream>>>(query, Wval, bval, logits, asum, Ubuf, v2w, hw, out);
}